// Ive_76553497084328
// MI455X (gfx1250) — compile-verified
//
#include <hip/hip_runtime.h>
#include <math.h>

// ---------------------------------------------------------------------------
// i1e(x) = exp(-|x|) * I1(x), single precision, Cephes/XLA coefficients.
// Memory-bound streaming kernel: 128-bit non-temporal loads/stores.
// ---------------------------------------------------------------------------

// Native clang vector type: required by __builtin_nontemporal_load/store.
typedef float v4f __attribute__((ext_vector_type(4)));

// Chebyshev coefficients for exp(-x) I1(x) / x on [0, 8]  (Cephes i1f.c, f32)
__device__ __constant__ float kI1eA[17] = {
     9.38153738649577178388e-9f,
    -4.44505912879632808065e-8f,
     2.00329475355213526229e-7f,
    -8.56872026469545474066e-7f,
     3.47025130813767847674e-6f,
    -1.32731636560394358279e-5f,
     4.78156510755005422638e-5f,
    -1.61760815825896745588e-4f,
     5.12285956168575772895e-4f,
    -1.51357245063125314899e-3f,
     4.15642294431288815669e-3f,
    -1.05640848946261981558e-2f,
     2.47264490306265168283e-2f,
    -5.29459812080949914269e-2f,
     1.02643658689847095384e-1f,
    -1.76416518357834055153e-1f,
     2.52587186443633654823e-1f
};

// Chebyshev coefficients for exp(-x) sqrt(x) I1(x) on (8, inf)  (Cephes, f32)
__device__ __constant__ float kI1eB[7] = {
    -3.83538038596423702205e-9f,
    -2.63146884688951950684e-8f,
    -2.51223623787020892529e-7f,
    -3.88256480887769039346e-6f,
    -1.10588938762623716291e-4f,
    -9.76109749136146840777e-3f,
     7.78576235018280120474e-1f
};

// Clenshaw recurrence: 0.5 * (b0 - b2) with b0 = y*b1 - b2 + c[i]
__device__ __forceinline__ float chbevl_17(float y) {
    float b0 = kI1eA[0];
    float b1 = 0.0f;
    float b2 = 0.0f;
#pragma unroll
    for (int i = 1; i < 17; ++i) {
        b2 = b1;
        b1 = b0;
        b0 = __fmaf_rn(y, b1, kI1eA[i] - b2);
    }
    return 0.5f * (b0 - b2);
}

__device__ __forceinline__ float chbevl_7(float y) {
    float b0 = kI1eB[0];
    float b1 = 0.0f;
    float b2 = 0.0f;
#pragma unroll
    for (int i = 1; i < 7; ++i) {
        b2 = b1;
        b1 = b0;
        b0 = __fmaf_rn(y, b1, kI1eB[i] - b2);
    }
    return 0.5f * (b0 - b2);
}

__device__ __forceinline__ float i1e_f32(float x) {
    const float z = fabsf(x);
    float r;
    if (z <= 8.0f) {
        // y in [-2, 2]
        const float y = __fmaf_rn(0.5f, z, -2.0f);
        r = chbevl_17(y) * z;
    } else {
        // Vanishingly rare for N(0,1) inputs; wave-uniformly skipped via execz.
        const float y = __fmaf_rn(32.0f, __frcp_rn(z), -2.0f);
        r = chbevl_7(y) * __frsqrt_rn(z);
    }
    // I1 is odd: i1e(-x) = -i1e(x). copysign also handles x == +-0 -> +-0.
    return copysignf(r, x);
}

// ---------------------------------------------------------------------------
// Kernels
// ---------------------------------------------------------------------------

__global__ void __launch_bounds__(256)
i1e_kernel_v4(const v4f* __restrict__ in, v4f* __restrict__ out,
              long long n4) {
    const long long i =
        (long long)blockIdx.x * (long long)blockDim.x + (long long)threadIdx.x;
    if (i < n4) {
        // Streaming data 2.8x larger than L2: non-temporal both ways.
        v4f v = __builtin_nontemporal_load(in + i);
        v4f r;
        r.x = i1e_f32(v.x);
        r.y = i1e_f32(v.y);
        r.z = i1e_f32(v.z);
        r.w = i1e_f32(v.w);
        __builtin_nontemporal_store(r, out + i);
    }
}

__global__ void __launch_bounds__(256)
i1e_kernel_tail(const float* __restrict__ in, float* __restrict__ out,
                long long base, long long n) {
    const long long i = base + (long long)blockIdx.x * (long long)blockDim.x +
                        (long long)threadIdx.x;
    if (i < n) {
        out[i] = i1e_f32(__builtin_nontemporal_load(in + i));
    }
}

// ---------------------------------------------------------------------------
// Entry point
// ---------------------------------------------------------------------------

extern "C" void kernel_launch(void* const* d_in, const int* in_sizes, int n_in,
                              void* d_out, int out_size, void* d_ws,
                              size_t ws_size, hipStream_t stream) {
    (void)n_in; (void)d_ws; (void)ws_size; (void)out_size;

    const float* z  = (const float*)d_in[0];
    float*       o  = (float*)d_out;
    const long long n  = (long long)in_sizes[0];
    const long long n4 = n >> 2;  // full float4s

    if (n4 > 0) {
        const int  block = 256;
        const long long grid = (n4 + block - 1) / block;
        i1e_kernel_v4<<<(unsigned int)grid, block, 0, stream>>>(
            (const v4f*)z, (v4f*)o, n4);
    }

    const long long rem = n - (n4 << 2);
    if (rem > 0) {
        i1e_kernel_tail<<<1, 256, 0, stream>>>(z, o, n4 << 2, n);
    }
}